// MultichannelMultiheadAttention_64974265254100
// MI455X (gfx1250) — compile-verified
//
#include <hip/hip_runtime.h>
#include <math.h>

typedef __attribute__((ext_vector_type(2))) float v2f;
typedef __attribute__((ext_vector_type(8))) float v8f;

#if defined(__HIP_DEVICE_COMPILE__)
#if !__has_builtin(__builtin_amdgcn_wmma_f32_16x16x4_f32)
#error "gfx1250 f32 WMMA builtin not available on this toolchain"
#endif
#endif

__device__ __forceinline__ v8f wmma4(v2f a, v2f b, v8f c) {
#if defined(__HIP_DEVICE_COMPILE__)
  // D = A(16x4,f32) * B(4x16,f32) + C(16x16,f32)  -> v_wmma_f32_16x16x4_f32
  return __builtin_amdgcn_wmma_f32_16x16x4_f32(false, a, false, b, (short)0, c,
                                               false, false);
#else
  (void)a; (void)b;
  return c;
#endif
}

// ---------------------------------------------------------------------------
// Problem constants: b=2, c=8, h=512, w=512(seq), heads=8, hd=64
// Z/Q/K/V buffers are laid out [b, c, feature(512), seq(512)] row-major.
// ---------------------------------------------------------------------------

// Channel mix for q/k/v: Y_p[b,d,h,w] = sum_c w_dw_p[d,c] * x[b,c,h,w]
__global__ void __launch_bounds__(256) mix3_kernel(
    const float* __restrict__ x, const float* __restrict__ wq,
    const float* __restrict__ wk, const float* __restrict__ wv,
    float* __restrict__ Yq, float* __restrict__ Yk, float* __restrict__ Yv) {
  unsigned i = blockIdx.x * 256u + threadIdx.x;  // over b*h*w = 524288
  unsigned b = i >> 18;
  unsigned hw = i & 262143u;
  size_t base = (size_t)b * 8 * 262144 + hw;
  float xv[8];
#pragma unroll
  for (int c = 0; c < 8; ++c) xv[c] = x[base + (size_t)c * 262144];
#pragma unroll
  for (int d = 0; d < 8; ++d) {
    float aq = 0.f, ak = 0.f, av = 0.f;
#pragma unroll
    for (int c = 0; c < 8; ++c) {
      aq += wq[d * 8 + c] * xv[c];
      ak += wk[d * 8 + c] * xv[c];
      av += wv[d * 8 + c] * xv[c];
    }
    size_t o = base + (size_t)d * 262144;
    Yq[o] = aq; Yk[o] = ak; Yv[o] = av;
  }
}

// Channel mix for output projection
__global__ void __launch_bounds__(256) mix1_kernel(
    const float* __restrict__ A, const float* __restrict__ wdw,
    float* __restrict__ Yo) {
  unsigned i = blockIdx.x * 256u + threadIdx.x;
  unsigned b = i >> 18;
  unsigned hw = i & 262143u;
  size_t base = (size_t)b * 8 * 262144 + hw;
  float xv[8];
#pragma unroll
  for (int c = 0; c < 8; ++c) xv[c] = A[base + (size_t)c * 262144];
#pragma unroll
  for (int d = 0; d < 8; ++d) {
    float s = 0.f;
#pragma unroll
    for (int c = 0; c < 8; ++c) s += wdw[d * 8 + c] * xv[c];
    Yo[base + (size_t)d * 262144] = s;
  }
}

// Per-(b,c) GEMM: Z[g,s] = sum_h Wp[c][g,h] * Y[b,c][h,s]   (512x512x512)
// Workgroup tile 64x64, BK=16; 8 waves, each wave owns one 16-row block and
// two 16x16 WMMA accumulators (32 cols).
__global__ void __launch_bounds__(256) gemm_pw_kernel(
    const float* __restrict__ Wp, const float* __restrict__ Y,
    float* __restrict__ Z) {
  const int bc = blockIdx.z, ch = bc & 7;
  const float* A = Wp + (size_t)ch * 262144;  // [g][h]
  const float* B = Y + (size_t)bc * 262144;   // [h][s]
  float* C = Z + (size_t)bc * 262144;
  const int m0 = blockIdx.y * 64, n0 = blockIdx.x * 64;
  __shared__ float As[64][20];  // [m][k], pad 20 for align/banks
  __shared__ float Bs[64][20];  // [n][k] (transposed)
  const int tid = threadIdx.x, lane = tid & 31, wvi = tid >> 5;
  const int wm = (wvi & 3) * 16, wn = (wvi >> 2) * 32;
  const int fml = wm + (lane & 15);
  const int fnl = wn + (lane & 15);
  const int fko = 2 * (lane >> 4);
  v8f acc0 = {}, acc1 = {};
  for (int k0 = 0; k0 < 512; k0 += 16) {
    {
      int r = tid >> 2, c4 = (tid & 3) * 4;  // A: 64 rows x 16 cols
      float4 va = *(const float4*)(A + (size_t)(m0 + r) * 512 + k0 + c4);
      *(float4*)&As[r][c4] = va;
    }
    {
      int r = tid >> 4, c4 = (tid & 15) * 4;  // B: 16 rows x 64 cols
      float4 vb = *(const float4*)(B + (size_t)(k0 + r) * 512 + n0 + c4);
      Bs[c4 + 0][r] = vb.x; Bs[c4 + 1][r] = vb.y;
      Bs[c4 + 2][r] = vb.z; Bs[c4 + 3][r] = vb.w;
    }
    __syncthreads();
#pragma unroll
    for (int kb = 0; kb < 16; kb += 4) {
      v2f a  = *(const v2f*)&As[fml][kb + fko];
      v2f b0 = *(const v2f*)&Bs[fnl][kb + fko];
      v2f b1 = *(const v2f*)&Bs[fnl + 16][kb + fko];
      acc0 = wmma4(a, b0, acc0);
      acc1 = wmma4(a, b1, acc1);
    }
    __syncthreads();
  }
  const int col = lane & 15, rb = 8 * (lane >> 4);
#pragma unroll
  for (int j = 0; j < 8; ++j) {
    int m = m0 + wm + rb + j;
    C[(size_t)m * 512 + n0 + wn + col] = acc0[j];
    C[(size_t)m * 512 + n0 + wn + 16 + col] = acc1[j];
  }
}

// In-place rotary on Q and K, layout [row=feature][seq]; pair rows (2i,2i+1)
__global__ void __launch_bounds__(256) rotary_kernel(float* __restrict__ Q,
                                                     float* __restrict__ K) {
  unsigned i = blockIdx.x * 256u + threadIdx.x;  // 2,097,152 = 4096 pairs * 512
  int s = i & 511;
  unsigned gp = i >> 9;     // global pair-row index
  int ip = gp & 31;         // pair index within head (d = 2*ip, hd=64)
  float freq = 1.0f / powf(10000.0f, (float)(2 * ip) / 64.0f);
  float ang = (float)s * freq;
  float sn, cs;
  sincosf(ang, &sn, &cs);
  size_t i0 = (size_t)gp * 1024 + s;  // row 2*gp
  size_t i1 = i0 + 512;               // row 2*gp+1
  float q0 = Q[i0], q1 = Q[i1];
  Q[i0] = q0 * cs - q1 * sn;
  Q[i1] = q1 * cs + q0 * sn;
  float k0 = K[i0], k1 = K[i1];
  K[i0] = k0 * cs - k1 * sn;
  K[i1] = k1 * cs + k0 * sn;
}

// Per-head: qk[s,t] = (q . k)/sqrt(512) + prev_qk ; q,k stored [d][seq]
__global__ void __launch_bounds__(256) gemm_qk_kernel(
    const float* __restrict__ Qz, const float* __restrict__ Kz,
    const float* __restrict__ prev, float* __restrict__ qk) {
  const int z = blockIdx.z, bc = z >> 3, hidx = z & 7;
  const float* Aq = Qz + (size_t)bc * 262144 + (size_t)hidx * 64 * 512;
  const float* Bk = Kz + (size_t)bc * 262144 + (size_t)hidx * 64 * 512;
  const size_t obase = (size_t)z * 262144;
  const int m0 = blockIdx.y * 64, n0 = blockIdx.x * 64;
  __shared__ float As[64][20];  // [s_local][d]
  __shared__ float Bs[64][20];  // [t_local][d]
  const int tid = threadIdx.x, lane = tid & 31, wvi = tid >> 5;
  const int wm = (wvi & 3) * 16, wn = (wvi >> 2) * 32;
  const int fml = wm + (lane & 15), fnl = wn + (lane & 15);
  const int fko = 2 * (lane >> 4);
  v8f acc0 = {}, acc1 = {};
  for (int k0 = 0; k0 < 64; k0 += 16) {
    int kr = tid >> 4, mc = (tid & 15) * 4;  // 16 d-rows x 64 cols
    float4 va = *(const float4*)(Aq + (size_t)(k0 + kr) * 512 + m0 + mc);
    As[mc + 0][kr] = va.x; As[mc + 1][kr] = va.y;
    As[mc + 2][kr] = va.z; As[mc + 3][kr] = va.w;
    float4 vb = *(const float4*)(Bk + (size_t)(k0 + kr) * 512 + n0 + mc);
    Bs[mc + 0][kr] = vb.x; Bs[mc + 1][kr] = vb.y;
    Bs[mc + 2][kr] = vb.z; Bs[mc + 3][kr] = vb.w;
    __syncthreads();
#pragma unroll
    for (int kb = 0; kb < 16; kb += 4) {
      v2f a  = *(const v2f*)&As[fml][kb + fko];
      v2f b0 = *(const v2f*)&Bs[fnl][kb + fko];
      v2f b1 = *(const v2f*)&Bs[fnl + 16][kb + fko];
      acc0 = wmma4(a, b0, acc0);
      acc1 = wmma4(a, b1, acc1);
    }
    __syncthreads();
  }
  const float scale = 0.044194173824159216f;  // 1/sqrt(512)
  const int col = lane & 15, rb = 8 * (lane >> 4);
#pragma unroll
  for (int j = 0; j < 8; ++j) {
    int m = m0 + wm + rb + j;
    size_t i0 = obase + (size_t)m * 512 + n0 + wn + col;
    size_t i1 = i0 + 16;
    qk[i0] = acc0[j] * scale + prev[i0];
    qk[i1] = acc1[j] * scale + prev[i1];
  }
}

// Row-wise softmax stats: per row of qk store (max, 1/sum(exp(.-max)))
__global__ void __launch_bounds__(256) softmax_stats_kernel(
    const float* __restrict__ qk, float* __restrict__ mstat,
    float* __restrict__ zstat) {
  int row = blockIdx.x * 8 + (threadIdx.x >> 5);  // 8 waves per block
  int lane = threadIdx.x & 31;
  const float* r = qk + (size_t)row * 512;
  float v[16];
  float mx = -INFINITY;
#pragma unroll
  for (int i = 0; i < 16; ++i) {
    v[i] = r[lane + i * 32];
    mx = fmaxf(mx, v[i]);
  }
#pragma unroll
  for (int o = 16; o; o >>= 1) mx = fmaxf(mx, __shfl_xor(mx, o, 32));
  float s = 0.f;
#pragma unroll
  for (int i = 0; i < 16; ++i) s += expf(v[i] - mx);
#pragma unroll
  for (int o = 16; o; o >>= 1) s += __shfl_xor(s, o, 32);
  if (lane == 0) {
    mstat[row] = mx;
    zstat[row] = 1.0f / s;
  }
}

// Per-head PV in merged layout: Az[d, s] = sum_t V[d,t] * softmax(qk)[s,t]
__global__ void __launch_bounds__(256) gemm_pv_kernel(
    const float* __restrict__ Vz, const float* __restrict__ qk,
    const float* __restrict__ mstat, const float* __restrict__ zstat,
    float* __restrict__ Az) {
  const int z = blockIdx.z, bc = z >> 3, hidx = z & 7;
  const float* Av = Vz + (size_t)bc * 262144 + (size_t)hidx * 64 * 512;  // [d][t]
  const float* P = qk + (size_t)z * 262144;                              // [s][t]
  const float* ms = mstat + (size_t)z * 512;
  const float* zs = zstat + (size_t)z * 512;
  float* Cz = Az + (size_t)bc * 262144 + (size_t)hidx * 64 * 512;
  const int n0 = blockIdx.x * 64;  // s block; M = 64 (all of d)
  __shared__ float As[64][20];     // [d][t_local]
  __shared__ float Bs[64][20];     // [s_local][t_local] = p
  const int tid = threadIdx.x, lane = tid & 31, wvi = tid >> 5;
  const int wm = (wvi & 3) * 16, wn = (wvi >> 2) * 32;
  const int fml = wm + (lane & 15), fnl = wn + (lane & 15);
  const int fko = 2 * (lane >> 4);
  v8f acc0 = {}, acc1 = {};
  for (int k0 = 0; k0 < 512; k0 += 16) {
    int r = tid >> 2, c4 = (tid & 3) * 4;  // 64 rows x 16 cols each tile
    float4 va = *(const float4*)(Av + (size_t)r * 512 + k0 + c4);
    *(float4*)&As[r][c4] = va;
    float4 vq = *(const float4*)(P + (size_t)(n0 + r) * 512 + k0 + c4);
    float mx = ms[n0 + r], rz = zs[n0 + r];
    float4 pb;
    pb.x = expf(vq.x - mx) * rz;
    pb.y = expf(vq.y - mx) * rz;
    pb.z = expf(vq.z - mx) * rz;
    pb.w = expf(vq.w - mx) * rz;
    *(float4*)&Bs[r][c4] = pb;
    __syncthreads();
#pragma unroll
    for (int kb = 0; kb < 16; kb += 4) {
      v2f a  = *(const v2f*)&As[fml][kb + fko];
      v2f b0 = *(const v2f*)&Bs[fnl][kb + fko];
      v2f b1 = *(const v2f*)&Bs[fnl + 16][kb + fko];
      acc0 = wmma4(a, b0, acc0);
      acc1 = wmma4(a, b1, acc1);
    }
    __syncthreads();
  }
  const int col = lane & 15, rb = 8 * (lane >> 4);
#pragma unroll
  for (int j = 0; j < 8; ++j) {
    int m = wm + rb + j;  // d
    Cz[(size_t)m * 512 + n0 + wn + col] = acc0[j];
    Cz[(size_t)m * 512 + n0 + wn + 16 + col] = acc1[j];
  }
}

extern "C" void kernel_launch(void* const* d_in, const int* in_sizes, int n_in,
                              void* d_out, int out_size, void* d_ws,
                              size_t ws_size, hipStream_t stream) {
  (void)in_sizes; (void)n_in; (void)out_size; (void)ws_size;
  const float* x       = (const float*)d_in[0];
  const float* prev_qk = (const float*)d_in[1];
  const float* wq_dw   = (const float*)d_in[2];
  const float* wq_pw   = (const float*)d_in[3];
  const float* wk_dw   = (const float*)d_in[4];
  const float* wk_pw   = (const float*)d_in[5];
  const float* wv_dw   = (const float*)d_in[6];
  const float* wv_pw   = (const float*)d_in[7];
  const float* wo_dw   = (const float*)d_in[8];
  const float* wo_pw   = (const float*)d_in[9];

  float* out = (float*)d_out;          // (2,8,512,512) = 4,194,304 floats
  float* qk_out = out + 4194304;       // (2,8,8,512,512) = 33,554,432 floats

  const size_t N1 = 4194304;           // one (b,c,512,512) buffer
  float* ws = (float*)d_ws;
  float* Yq = ws + 0 * N1;
  float* Yk = ws + 1 * N1;
  float* Yv = ws + 2 * N1;
  float* Qz = ws + 3 * N1;
  float* Kz = ws + 4 * N1;
  float* Vz = ws + 5 * N1;
  float* Az = Yq;                      // reuse: Yq dead after Qz GEMM
  float* Yo = Yk;                      // reuse: Yk dead after Kz GEMM
  float* mstat = Yv;                   // reuse: Yv dead after Vz GEMM (65536)
  float* zstat = Yv + 65536;

  dim3 blk(256);
  dim3 gPW(8, 8, 16);

  mix3_kernel<<<2048, blk, 0, stream>>>(x, wq_dw, wk_dw, wv_dw, Yq, Yk, Yv);
  gemm_pw_kernel<<<gPW, blk, 0, stream>>>(wq_pw, Yq, Qz);
  gemm_pw_kernel<<<gPW, blk, 0, stream>>>(wk_pw, Yk, Kz);
  gemm_pw_kernel<<<gPW, blk, 0, stream>>>(wv_pw, Yv, Vz);
  rotary_kernel<<<8192, blk, 0, stream>>>(Qz, Kz);
  gemm_qk_kernel<<<dim3(8, 8, 128), blk, 0, stream>>>(Qz, Kz, prev_qk, qk_out);
  softmax_stats_kernel<<<8192, blk, 0, stream>>>(qk_out, mstat, zstat);
  gemm_pv_kernel<<<dim3(8, 1, 128), blk, 0, stream>>>(Vz, qk_out, mstat, zstat, Az);
  mix1_kernel<<<2048, blk, 0, stream>>>(Az, wo_dw, Yo);
  gemm_pw_kernel<<<gPW, blk, 0, stream>>>(wo_pw, Yo, out);
}